// SparseLinear2_26018911879781
// MI455X (gfx1250) — compile-verified
//
#include <hip/hip_runtime.h>

// ---------------------------------------------------------------------------
// SparseLinear2 on MI455X (gfx1250, wave32)
//
//   out[b,m] = sum_{e: dst[e]==m} x[b, src[e]] * values[e]  + bias[m]
//
// Pipeline (fast path, B == 16):
//   1) k_zero        : zero acc (M x 16) in workspace
//   2) k_transpose16 : xt (N x 16) = x (16 x N)^T   -> 64B rows per node
//   3) k_edges16     : async-stage 256 edges/block into LDS, then
//                      16 lanes/edge gather xt[src] row (64B coalesced) and
//                      global_atomic_add_f32 into acc[dst] row
//   4) k_finalize_wmma: per 16x16 tile, D = A x I + C via 4x
//                      v_wmma_f32_16x16x4_f32 (A = acc tile, C = bias
//                      broadcast), stored as out[b, m] (64B segments)
// ---------------------------------------------------------------------------

#define THREADS 256

typedef __attribute__((ext_vector_type(2))) float v2f;
typedef __attribute__((ext_vector_type(8))) float v8f;

#if defined(__HIP_DEVICE_COMPILE__)
typedef __attribute__((address_space(1))) int g_i32;  // global
typedef __attribute__((address_space(3))) int l_i32;  // LDS
#endif

__device__ __forceinline__ void atomic_add_f32(float* p, float v) {
  // Hardware global_atomic_add_f32 (relaxed, device scope).
  __hip_atomic_fetch_add(p, v, __ATOMIC_RELAXED, __HIP_MEMORY_SCOPE_AGENT);
}

// --------------------------- utility kernels -------------------------------

__global__ void k_zero4(float4* __restrict__ p, int n4) {
  int i = blockIdx.x * blockDim.x + threadIdx.x;
  if (i < n4) p[i] = make_float4(0.f, 0.f, 0.f, 0.f);
}

__global__ void k_zero1(float* __restrict__ p, int n) {
  int i = blockIdx.x * blockDim.x + threadIdx.x;
  if (i < n) p[i] = 0.f;
}

// xt[n*16 + b] = x[b*N + n]   (B == 16). Coalesced reads, 64B-strided writes.
__global__ void k_transpose16(const float* __restrict__ x, float* __restrict__ xt,
                              int N, int total) {
  int t = blockIdx.x * blockDim.x + threadIdx.x;
  if (t >= total) return;
  int b = t / N;
  int n = t - b * N;
  xt[n * 16 + b] = x[t];
}

// --------------------------- edge scatter ----------------------------------

__global__ void k_edges16(const int* __restrict__ idx, const float* __restrict__ vals,
                          const float* __restrict__ xt, float* __restrict__ acc,
                          int E) {
  __shared__ int   s_src[THREADS];
  __shared__ int   s_dst[THREADS];
  __shared__ float s_val[THREADS];

  const int t  = threadIdx.x;
  const int e0 = blockIdx.x * THREADS;
  const int e  = e0 + t;

#if defined(__HIP_DEVICE_COMPILE__) &&                               \
    __has_builtin(__builtin_amdgcn_global_load_async_to_lds_b32) &&  \
    __has_builtin(__builtin_amdgcn_s_wait_asynccnt)
  // CDNA5 async global->LDS staging of the edge stream (ASYNCcnt tracked).
  if (e < E) {
    __builtin_amdgcn_global_load_async_to_lds_b32(
        (g_i32*)(idx + e), (l_i32*)&s_src[t], 0, 0);
    __builtin_amdgcn_global_load_async_to_lds_b32(
        (g_i32*)(idx + E + e), (l_i32*)&s_dst[t], 0, 0);
    __builtin_amdgcn_global_load_async_to_lds_b32(
        (g_i32*)(vals + e), (l_i32*)&s_val[t], 0, 0);
  }
  __builtin_amdgcn_s_wait_asynccnt(0);
  __syncthreads();
#else
  if (e < E) {
    s_src[t] = idx[e];
    s_dst[t] = idx[E + e];
    s_val[t] = vals[e];
  }
  __syncthreads();
#endif

  const int b = t & 15;  // batch lane
  const int g = t >> 4;  // edge group (16 groups of 16 lanes)

#pragma unroll
  for (int i = 0; i < 16; ++i) {
    int el = g * 16 + i;
    if (e0 + el >= E) break;
    int   s = s_src[el];
    int   d = s_dst[el];
    float v = s_val[el];
    // lanes 0..15 of each group read 64B contiguous, add 64B contiguous
    atomic_add_f32(&acc[(size_t)d * 16 + b], xt[(size_t)s * 16 + b] * v);
  }
}

// --------------------------- WMMA finalize ---------------------------------
// One wave per 16x16 tile:  out[b, m0+j] = acc[(m0+j)*16 + b] + bias[m0+j]
// Computed as D = A x I + C with V_WMMA_F32_16X16X4_F32 chained over K.
//   A(m=b, k) = acc[(m0+k)*16 + b]   (A-layout: lane%16 = m, vgpr/lanehi = k)
//   B = identity slices (4x16)       (B-layout: lane%16 = n, vgpr/lanehi = k)
//   C(m, n)   = bias[m0 + n]
//   D(m=b, n=j) stored to out[b*M + m0 + j]
__global__ void k_finalize_wmma(const float* __restrict__ acc,
                                const float* __restrict__ bias,
                                float* __restrict__ out,
                                int M, int tiles) {
  int wave = (blockIdx.x * blockDim.x + threadIdx.x) >> 5;  // global wave id
  int lane = threadIdx.x & 31;
  if (wave >= tiles) return;  // wave-uniform exit: EXEC stays all-1s for WMMA

  const int m0  = wave * 16;
  const int col = lane & 15;  // n (and A's m) within tile
  const int hi  = lane >> 4;  // lane half selects K pair (2*hi, 2*hi+1)

  const float bv = bias[m0 + col];
  v8f d = {bv, bv, bv, bv, bv, bv, bv, bv};  // C = bias broadcast per column

#pragma unroll
  for (int ch = 0; ch < 4; ++ch) {
    const int kg = ch * 4 + hi * 2;  // global K index of this lane's first elem
    v2f a;
    a.x = acc[(size_t)(m0 + kg) * 16 + col];        // A(col, kg)
    a.y = acc[(size_t)(m0 + kg + 1) * 16 + col];    // A(col, kg+1)
    v2f bm;
    bm.x = (col == kg) ? 1.0f : 0.0f;               // I(kg,   col)
    bm.y = (col == kg + 1) ? 1.0f : 0.0f;           // I(kg+1, col)
    d = __builtin_amdgcn_wmma_f32_16x16x4_f32(
        /*neg_a=*/false, a, /*neg_b=*/false, bm,
        /*c_mod=*/(short)0, d, /*reuse_a=*/false, /*reuse_b=*/false);
  }

#pragma unroll
  for (int r = 0; r < 8; ++r) {
    int b = r + hi * 8;  // D row = batch index
    out[(size_t)b * M + m0 + col] = d[r];
  }
}

// scalar tail for M not divisible by 16 (not hit for M=50000)
__global__ void k_finalize_tail(const float* __restrict__ acc,
                                const float* __restrict__ bias,
                                float* __restrict__ out,
                                int M, int mStart, int count /* = (M-mStart)*16 */) {
  int t = blockIdx.x * blockDim.x + threadIdx.x;
  if (t >= count) return;
  int m = mStart + t / 16;
  int b = t & 15;
  out[(size_t)b * M + m] = acc[(size_t)m * 16 + b] + bias[m];
}

// --------------------------- generic fallback ------------------------------

__global__ void k_edges_generic(const int* __restrict__ idx, const float* __restrict__ vals,
                                const float* __restrict__ x, float* __restrict__ out,
                                int E, int B, int N, int M) {
  long long t = (long long)blockIdx.x * blockDim.x + threadIdx.x;
  long long total = (long long)E * B;
  if (t >= total) return;
  int e = (int)(t / B);
  int b = (int)(t - (long long)e * B);
  int s = idx[e];
  int d = idx[E + e];
  atomic_add_f32(&out[(size_t)b * M + d], x[(size_t)b * N + s] * vals[e]);
}

__global__ void k_bias_add(float* __restrict__ out, const float* __restrict__ bias,
                           int M, int total) {
  int t = blockIdx.x * blockDim.x + threadIdx.x;
  if (t >= total) return;
  out[t] += bias[t % M];
}

// --------------------------- launch ----------------------------------------

extern "C" void kernel_launch(void* const* d_in, const int* in_sizes, int n_in,
                              void* d_out, int out_size, void* d_ws, size_t ws_size,
                              hipStream_t stream) {
  const float* x    = (const float*)d_in[0];  // (B, N)
  const float* vals = (const float*)d_in[1];  // (E,)
  const float* bias = (const float*)d_in[2];  // (M,)
  const int*   idx  = (const int*)d_in[3];    // (2, E) int32

  const int E  = in_sizes[1];
  const int M  = in_sizes[2];
  const int BN = in_sizes[0];
  const int B  = (M > 0) ? (out_size / M) : 0;
  const int N  = (B > 0) ? (BN / B) : 0;
  float* out = (float*)d_out;

  const size_t need = ((size_t)N + (size_t)M) * 16 * sizeof(float);

  if (B == 16 && ws_size >= need) {
    float* xt  = (float*)d_ws;             // (N, 16)
    float* acc = xt + (size_t)N * 16;      // (M, 16)

    int n4 = (M * 16) / 4;
    k_zero4<<<(n4 + THREADS - 1) / THREADS, THREADS, 0, stream>>>((float4*)acc, n4);

    k_transpose16<<<(BN + THREADS - 1) / THREADS, THREADS, 0, stream>>>(x, xt, N, BN);

    k_edges16<<<(E + THREADS - 1) / THREADS, THREADS, 0, stream>>>(idx, vals, xt, acc, E);

    int tiles = M / 16;
    if (tiles > 0) {
      int nthreads = tiles * 32;
      k_finalize_wmma<<<(nthreads + THREADS - 1) / THREADS, THREADS, 0, stream>>>(
          acc, bias, out, M, tiles);
    }
    int rem = M - tiles * 16;
    if (rem > 0) {
      int count = rem * 16;
      k_finalize_tail<<<(count + THREADS - 1) / THREADS, THREADS, 0, stream>>>(
          acc, bias, out, M, tiles * 16, count);
    }
  } else {
    // generic path: accumulate directly into out, then add bias
    k_zero1<<<(out_size + THREADS - 1) / THREADS, THREADS, 0, stream>>>(out, out_size);
    long long total = (long long)E * B;
    long long blocks = (total + THREADS - 1) / THREADS;
    k_edges_generic<<<(unsigned)blocks, THREADS, 0, stream>>>(idx, vals, x, out, E, B, N, M);
    k_bias_add<<<(out_size + THREADS - 1) / THREADS, THREADS, 0, stream>>>(out, bias, M, out_size);
  }
}